// MSIWC2F_28535762714938
// MI455X (gfx1250) — compile-verified
//
#include <hip/hip_runtime.h>
#include <hip/hip_bf16.h>
#include <cstdint>

// Problem constants (fixed shape: nw_out [4, 19, 512, 1024] f32)
#define NB      4
#define CIN     19
#define CO      12
#define HW_     (512 * 1024)          // 524288 = 2^19 pixels per channel plane
#define NPIX    (NB * HW_)            // 2097152 = 2^21
#define TILE    256
#define NTILES  (NPIX / TILE)         // 8192
#define GRID1   1024                  // 8 tiles per block

// ---- CDNA5 async global->LDS path (ASYNCcnt) -------------------------------
#if defined(__gfx1250__) && __has_builtin(__builtin_amdgcn_global_load_async_to_lds_b32) && __has_builtin(__builtin_amdgcn_s_wait_asynccnt)
#define HAVE_ASYNC 1
#else
#define HAVE_ASYNC 0
#endif

#if HAVE_ASYNC
typedef __attribute__((address_space(1))) int as1_int;   // global (LangAS cuda_device)
typedef __attribute__((address_space(3))) int as3_int;   // LDS
// Global generic addresses == AS1 addresses on amdgcn; LDS flat address low 32
// bits == LDS offset (addrspacecast generic->AS3 is a 32-bit truncate).
#define TO_GLOBAL(p) ((as1_int*)(uintptr_t)(p))
#define TO_LDS(p)    ((as3_int*)(uint32_t)(uintptr_t)(p))
#endif

// Per-pixel work: softmax over 19 logits, 12 coarse planes, argmax + ssq,
// accumulate into LDS bins.
__device__ __forceinline__ void process_pixel(const float x[CIN],
                                              float* s_ssq, unsigned int* s_cnt) {
    float m = x[0];
#pragma unroll
    for (int c = 1; c < CIN; ++c) m = fmaxf(m, x[c]);
    float e[CIN];
    float s = 0.0f;
#pragma unroll
    for (int c = 0; c < CIN; ++c) { e[c] = __expf(x[c] - m); s += e[c]; }
    const float inv = 1.0f / s;

    // IDS_MAPPING = [[0],[1],[2,3,4],[5],[6,7],[8,9,10],[11],[12],[13,14,15],[16],[17],[18]]
    float o[CO];
    o[0]  = e[0] * inv;
    o[1]  = e[1] * inv;
    o[2]  = x[2] + x[3] + x[4];
    o[3]  = e[5] * inv;
    o[4]  = x[6] + x[7];
    o[5]  = x[8] + x[9] + x[10];
    o[6]  = e[11] * inv;
    o[7]  = e[12] * inv;
    o[8]  = x[13] + x[14] + x[15];
    o[9]  = e[16] * inv;
    o[10] = e[17] * inv;
    o[11] = e[18] * inv;

    float best = o[0];
    int   bi   = 0;
    float ssq  = o[0] * o[0];
#pragma unroll
    for (int c = 1; c < CO; ++c) {
        ssq += o[c] * o[c];
        if (o[c] > best) { best = o[c]; bi = c; }  // strict '>' keeps first max (jnp.argmax)
    }
    atomicAdd(&s_ssq[bi], ssq);      // ds_add_f32
    atomicAdd(&s_cnt[bi], 1u);       // ds_add_u32
}

__global__ void MSIWC2F_zero_ws(float* __restrict__ ssq, unsigned int* __restrict__ cnt) {
    const int i = threadIdx.x;
    if (i < CO) { ssq[i] = 0.0f; cnt[i] = 0u; }
}

__global__ __launch_bounds__(TILE) void MSIWC2F_pass1(const float* __restrict__ in,
                                                      float* __restrict__ g_ssq,
                                                      unsigned int* __restrict__ g_cnt) {
    __shared__ float        s_ssq[CO];
    __shared__ unsigned int s_cnt[CO];
    const int tid = threadIdx.x;
    if (tid < CO) { s_ssq[tid] = 0.0f; s_cnt[tid] = 0u; }
    __syncthreads();

    const int grid = gridDim.x;

#if HAVE_ASYNC
    __shared__ float buf[2][CIN * TILE];   // 2 x 19 x 256 x 4B = 38 KB of 320 KB/WGP

    // Issue one tile (19 wave-wide 128B async copies per wave; each lane copies
    // its own pixel's 19 channel values into its private LDS slots).
    auto issue = [&](int t, int stage) {
        const int    p   = t * TILE + tid;
        const float* gp  = in + (size_t)(p >> 19) * (size_t)(CIN * HW_) + (p & (HW_ - 1));
        float*       dst = &buf[stage][tid];
#pragma unroll
        for (int c = 0; c < CIN; ++c) {
            __builtin_amdgcn_global_load_async_to_lds_b32(
                TO_GLOBAL(gp + (size_t)c * HW_), TO_LDS(dst + c * TILE), 0, 0);
        }
    };

    int t0 = blockIdx.x;
    if (t0 < NTILES) issue(t0, 0);
    int stage = 0;
    for (int t = t0; t < NTILES; t += grid) {
        const int tn = t + grid;
        if (tn < NTILES) {
            issue(tn, stage ^ 1);                       // prefetch next tile
            __builtin_amdgcn_s_wait_asynccnt(CIN);      // oldest 19 (current tile) done
        } else {
            __builtin_amdgcn_s_wait_asynccnt(0);
        }
        asm volatile("" ::: "memory");
        float x[CIN];
#pragma unroll
        for (int c = 0; c < CIN; ++c) x[c] = buf[stage][c * TILE + tid];
        stage ^= 1;
        process_pixel(x, s_ssq, s_cnt);
    }
#else
    for (int t = blockIdx.x; t < NTILES; t += grid) {
        const int    p  = t * TILE + tid;
        const float* gp = in + (size_t)(p >> 19) * (size_t)(CIN * HW_) + (p & (HW_ - 1));
        float x[CIN];
#pragma unroll
        for (int c = 0; c < CIN; ++c) x[c] = gp[(size_t)c * HW_];
        process_pixel(x, s_ssq, s_cnt);
    }
#endif

    __syncthreads();
    if (tid < CO) {
        atomicAdd(&g_ssq[tid], s_ssq[tid]);
        atomicAdd(&g_cnt[tid], s_cnt[tid]);
    }
}

__global__ void MSIWC2F_finalize(const float* __restrict__ ssq,
                                 const unsigned int* __restrict__ cnt,
                                 float* __restrict__ out) {
    if (threadIdx.x == 0) {
        const float npw = powf((float)NPIX, 0.8f);   // Np^(1-iw)
        float tot = 0.0f;
        for (int k = 0; k < CO; ++k) {
            const float w = fmaxf(powf((float)cnt[k], 0.2f) * npw, 1.0f);
            tot += ssq[k] / w;
        }
        out[0] = -tot / (float)(NB * CO);            // /(N*C) = /48
    }
}

extern "C" void kernel_launch(void* const* d_in, const int* in_sizes, int n_in,
                              void* d_out, int out_size, void* d_ws, size_t ws_size,
                              hipStream_t stream) {
    (void)in_sizes; (void)n_in; (void)out_size; (void)ws_size;
    const float* in = (const float*)d_in[0];
    float*        ws_ssq = (float*)d_ws;
    unsigned int* ws_cnt = (unsigned int*)((float*)d_ws + 16);  // 64B-aligned region
    float*        out    = (float*)d_out;

    MSIWC2F_zero_ws<<<1, 32, 0, stream>>>(ws_ssq, ws_cnt);
    MSIWC2F_pass1<<<GRID1, TILE, 0, stream>>>(in, ws_ssq, ws_cnt);
    MSIWC2F_finalize<<<1, 32, 0, stream>>>(ws_ssq, ws_cnt, out);
}